// GraphDecoder_51771535786306
// MI455X (gfx1250) — compile-verified
//
#include <hip/hip_runtime.h>

// ---------------------------------------------------------------------------
// 2-layer GraphConv (DGL norm='both') for MI455X / gfx1250.
// Feature + agg buffers (25.6 MB each) are L2-resident (192 MB L2), so the
// edge gather/scatter runs at L2 rate; scatter-add must use the hardware
// GLOBAL_ATOMIC_ADD_F32 (unsafeAtomicAdd), not a CAS loop. GEMMs use
// V_WMMA_F32_16X16X4_F32 (exact f32 matrix path, wave32).
// ---------------------------------------------------------------------------

#define HIDDEN 64
#define OUT_F  30

typedef __attribute__((ext_vector_type(2))) float v2f;
typedef __attribute__((ext_vector_type(8))) float v8f;

// ----------------------------- utility kernels -----------------------------

__global__ void zero_f4(float4* __restrict__ p, int n4) {
    int i = blockIdx.x * blockDim.x + threadIdx.x;
    if (i < n4) p[i] = make_float4(0.f, 0.f, 0.f, 0.f);
}

__global__ void degree_k(const int* __restrict__ src, const int* __restrict__ dst,
                         float* __restrict__ deg_out, float* __restrict__ deg_in, int nE) {
    int e = blockIdx.x * blockDim.x + threadIdx.x;
    if (e < nE) {
        unsafeAtomicAdd(deg_out + src[e], 1.0f);
        unsafeAtomicAdd(deg_in  + dst[e], 1.0f);
    }
}

__global__ void norm_k(const float* __restrict__ deg_out, const float* __restrict__ deg_in,
                       float* __restrict__ norm_out, float* __restrict__ norm_in, int n) {
    int i = blockIdx.x * blockDim.x + threadIdx.x;
    if (i < n) {
        float d0 = deg_out[i];
        float d1 = deg_in[i];
        norm_out[i] = (d0 > 0.f) ? rsqrtf(d0) : 0.f;
        norm_in[i]  = (d1 > 0.f) ? rsqrtf(d1) : 0.f;
    }
}

// xs[node] = x[node] * norm_out[node]   (coalesced float4, 16 quads per node)
__global__ void scale_k(const float4* __restrict__ x4, const float* __restrict__ norm_out,
                        float4* __restrict__ xs4, int n4) {
    int i = blockIdx.x * blockDim.x + threadIdx.x;
    if (i < n4) {
        float s = norm_out[i >> 4];
        float4 v = x4[i];
        v.x *= s; v.y *= s; v.z *= s; v.w *= s;
        xs4[i] = v;
    }
}

// agg[dst] += feat[src]; one thread = (edge, 16-byte quad).
// float4 (b128) gather + 4 hardware f32 atomics (no-return -> STOREcnt only).
__global__ void scatter_k(const float4* __restrict__ feat4,
                          const int* __restrict__ src, const int* __restrict__ dst,
                          float* __restrict__ agg, int nE) {
    int gid = blockIdx.x * blockDim.x + threadIdx.x;
    int e = gid >> 4;
    int q = gid & 15;
    if (e < nE) {
        int s = src[e];
        int d = dst[e];
        float4 v = feat4[(size_t)s * 16 + q];
        float* o = agg + (size_t)d * 64 + q * 4;
        unsafeAtomicAdd(o + 0, v.x);
        unsafeAtomicAdd(o + 1, v.y);
        unsafeAtomicAdd(o + 2, v.z);
        unsafeAtomicAdd(o + 3, v.w);
    }
}

// ------------------------------- WMMA GEMMs --------------------------------
// Layer 1: h1s = relu((agg * norm_in) @ W1 + b1) * norm_out    [N x 64]
// One wave per 16-node tile. A fragments (16 k-steps) cached in registers,
// loop over 4 column tiles of 16.
__global__ void gemm1_k(const float* __restrict__ agg, const float* __restrict__ norm_in,
                        const float* __restrict__ norm_out,
                        const float* __restrict__ W1, const float* __restrict__ b1,
                        float* __restrict__ h1s, int ntiles) {
    int wave = blockIdx.x * (blockDim.x >> 5) + (threadIdx.x >> 5);
    if (wave >= ntiles) return;              // wave-uniform: EXEC stays all-ones
    int lane  = threadIdx.x & 31;
    int m     = lane & 15;                   // A row / B column within tile
    int khalf = lane >> 4;                   // K sub-pair select
    int r0    = wave * 16;
    int row   = r0 + m;

    float nin = norm_in[row];
    // A fragments: lane holds A[row][4k + 2*khalf + {0,1}] for k = 0..15
    v2f a[16];
    const float2* ap2 = reinterpret_cast<const float2*>(agg + (size_t)row * 64 + khalf * 2);
#pragma unroll
    for (int k = 0; k < 16; ++k) {
        float2 t = ap2[2 * k];
        a[k].x = t.x * nin;
        a[k].y = t.y * nin;
    }

    float nout[8];
#pragma unroll
    for (int r = 0; r < 8; ++r) nout[r] = norm_out[r0 + khalf * 8 + r];

#pragma unroll
    for (int nt = 0; nt < 4; ++nt) {
        int n = nt * 16 + m;
        float bias = b1[n];
        v8f c;
#pragma unroll
        for (int r = 0; r < 8; ++r) c[r] = bias;
#pragma unroll
        for (int k = 0; k < 16; ++k) {
            int kk = k * 4 + khalf * 2;
            v2f b;
            b.x = W1[kk * HIDDEN + n];
            b.y = W1[(kk + 1) * HIDDEN + n];
            c = __builtin_amdgcn_wmma_f32_16x16x4_f32(
                    false, a[k], false, b, (short)0, c, false, false);
        }
#pragma unroll
        for (int r = 0; r < 8; ++r) {
            int rown = r0 + khalf * 8 + r;
            float v = c[r];
            v = (v > 0.f) ? v : 0.f;         // ReLU
            h1s[(size_t)rown * 64 + n] = v * nout[r];  // pre-scale for layer 2
        }
    }
}

// Layer 2: out = (agg * norm_in) @ W2 + b2   [N x 30], W2 padded to 32 cols.
__global__ void gemm2_k(const float* __restrict__ agg, const float* __restrict__ norm_in,
                        const float* __restrict__ W2, const float* __restrict__ b2,
                        float* __restrict__ out, int ntiles) {
    int wave = blockIdx.x * (blockDim.x >> 5) + (threadIdx.x >> 5);
    if (wave >= ntiles) return;
    int lane  = threadIdx.x & 31;
    int m     = lane & 15;
    int khalf = lane >> 4;
    int r0    = wave * 16;
    int row   = r0 + m;

    float nin = norm_in[row];
    v2f a[16];
    const float2* ap2 = reinterpret_cast<const float2*>(agg + (size_t)row * 64 + khalf * 2);
#pragma unroll
    for (int k = 0; k < 16; ++k) {
        float2 t = ap2[2 * k];
        a[k].x = t.x * nin;
        a[k].y = t.y * nin;
    }

#pragma unroll
    for (int nt = 0; nt < 2; ++nt) {
        int n = nt * 16 + m;
        int nc = (n < OUT_F) ? n : 0;        // clamp (branch-free)
        float msk = (n < OUT_F) ? 1.f : 0.f;
        float bias = b2[nc] * msk;
        v8f c;
#pragma unroll
        for (int r = 0; r < 8; ++r) c[r] = bias;
#pragma unroll
        for (int k = 0; k < 16; ++k) {
            int kk = k * 4 + khalf * 2;
            v2f b;
            b.x = W2[kk * OUT_F + nc] * msk;
            b.y = W2[(kk + 1) * OUT_F + nc] * msk;
            c = __builtin_amdgcn_wmma_f32_16x16x4_f32(
                    false, a[k], false, b, (short)0, c, false, false);
        }
        if (n < OUT_F) {
#pragma unroll
            for (int r = 0; r < 8; ++r) {
                int rown = r0 + khalf * 8 + r;
                out[(size_t)rown * OUT_F + n] = c[r];
            }
        }
    }
}

// ------------------------------- host launch -------------------------------

extern "C" void kernel_launch(void* const* d_in, const int* in_sizes, int n_in,
                              void* d_out, int out_size, void* d_ws, size_t ws_size,
                              hipStream_t stream) {
    const float* x   = (const float*)d_in[0];
    const int*   src = (const int*)  d_in[1];
    const int*   dst = (const int*)  d_in[2];
    const float* W1  = (const float*)d_in[3];
    const float* b1  = (const float*)d_in[4];
    const float* W2  = (const float*)d_in[5];
    const float* b2  = (const float*)d_in[6];
    float* out = (float*)d_out;

    const int nN = in_sizes[0] / HIDDEN;   // 100000
    const int nE = in_sizes[1];            // 3200000

    // Workspace layout (floats): deg_out|deg_in|norm_out|norm_in|aggbuf|featbuf
    float* ws       = (float*)d_ws;
    float* deg_out  = ws;
    float* deg_in   = ws + (size_t)nN;
    float* norm_out = ws + (size_t)2 * nN;
    float* norm_in  = ws + (size_t)3 * nN;
    float* aggb     = ws + (size_t)4 * nN;                 // [N x 64] (reused both layers)
    float* feat     = aggb + (size_t)nN * HIDDEN;          // [N x 64] (xs, then h1s)

    const int ntiles = nN / 16;            // 6250 exact
    const int TB = 256;

    // 1) zero degrees + agg
    zero_f4<<<((2 * nN / 4) + TB - 1) / TB, TB, 0, stream>>>((float4*)deg_out, 2 * nN / 4);
    zero_f4<<<((nN * 16) + TB - 1) / TB, TB, 0, stream>>>((float4*)aggb, nN * 16);

    // 2) degrees
    degree_k<<<(nE + TB - 1) / TB, TB, 0, stream>>>(src, dst, deg_out, deg_in, nE);

    // 3) norms
    norm_k<<<(nN + TB - 1) / TB, TB, 0, stream>>>(deg_out, deg_in, norm_out, norm_in, nN);

    // 4) xs = x * norm_out
    scale_k<<<((nN * 16) + TB - 1) / TB, TB, 0, stream>>>((const float4*)x, norm_out,
                                                          (float4*)feat, nN * 16);

    // 5) scatter layer 1: agg += xs[src] at dst
    {
        long long nthr = (long long)nE * 16;
        scatter_k<<<(int)((nthr + TB - 1) / TB), TB, 0, stream>>>(
            (const float4*)feat, src, dst, aggb, nE);
    }

    // 6) GEMM1 (WMMA f32): feat <- relu((agg*norm_in)@W1 + b1) * norm_out
    gemm1_k<<<(ntiles + 3) / 4, 128, 0, stream>>>(aggb, norm_in, norm_out, W1, b1, feat, ntiles);

    // 7) zero agg, scatter layer 2
    zero_f4<<<((nN * 16) + TB - 1) / TB, TB, 0, stream>>>((float4*)aggb, nN * 16);
    {
        long long nthr = (long long)nE * 16;
        scatter_k<<<(int)((nthr + TB - 1) / TB), TB, 0, stream>>>(
            (const float4*)feat, src, dst, aggb, nE);
    }

    // 8) GEMM2 (WMMA f32): out = (agg*norm_in)@W2 + b2
    gemm2_k<<<(ntiles + 3) / 4, 128, 0, stream>>>(aggb, norm_in, W2, b2, out, ntiles);
}